// BiomechanicsLoss_85916525789529
// MI455X (gfx1250) — compile-verified
//
#include <hip/hip_runtime.h>
#include <math.h>

typedef __attribute__((ext_vector_type(2))) float v2f;
typedef __attribute__((ext_vector_type(8))) float v8f;

// Stiffness matrix C = inv(Ci) (computed exactly from the reference's
// compliance matrix, including its deliberate asymmetry). Block structure:
//   [ C00 C01 C02 ]                 C00 = 151/555, C01 = 271/2220,
//   [ C01 C00 C02 ]   (top-left),   C02 = 35/222,  C20 = 343/10360,
//   [ C20 C20 C22 ]                 C22 = 245/1036, shear diag = 0.075
#define C00 0.27207207f
#define C01 0.12207207f
#define C02 0.15765766f
#define C20 0.03310811f
#define C22 0.23648649f
#define CSH 0.075f

__global__ void biomech_init_ws(float* ws) {
    if (threadIdx.x == 0) { ws[0] = 0.0f; ws[1] = 0.0f; }
}

__global__ __launch_bounds__(256) void biomech_wmma_kernel(
    const float* __restrict__ gu, const float* __restrict__ gv,
    const float* __restrict__ gw, const float* __restrict__ sdf,
    float* __restrict__ ws, int n)
{
    const int lane          = threadIdx.x & 31;          // wave32
    const int wavesPerBlock = blockDim.x >> 5;
    const int waveId        = blockIdx.x * wavesPerBlock + (threadIdx.x >> 5);
    const int nWaves        = gridDim.x * wavesPerBlock;
    const int nTiles        = (n + 15) >> 4;             // 16 points per tile

    const bool lower = (lane < 16);

    float sumsq = 0.0f;   // per-lane partial of sum(q^2)
    float cnt   = 0.0f;   // per-lane partial of mask count (lower half only)

    for (int tile = waveId; tile < nTiles; tile += nWaves) {
        const int p   = (tile << 4) + (lane & 15);
        const bool ok = (p < n);
        const int pi  = ok ? p : 0;

        // Coalesced loads: consecutive lanes read consecutive 12B rows -> b96.
        const float* __restrict__ up = gu + 3 * pi;
        const float* __restrict__ vp = gv + 3 * pi;
        const float* __restrict__ wp = gw + 3 * pi;
        const float u0 = up[0], u1 = up[1], u2 = up[2];
        const float v0 = vp[0], v1 = vp[1], v2 = vp[2];
        const float w0 = wp[0], w1 = wp[1], w2 = wp[2];
        const float s  = sdf[pi];

        // Fold sdf mask into the strain: masked-out points give q == 0, so
        // the WMMA needs no predication (EXEC must stay all-ones).
        const float m  = (ok && (s < 1e-8f)) ? 1.0f : 0.0f;
        const float e0 = u0 * m;
        const float e1 = v1 * m;
        const float e2 = w2 * m;
        const float e3 = 0.5f * (u1 + v0) * m;
        const float e4 = 0.5f * (u2 + w0) * m;
        const float e5 = 0.5f * (w1 + v2) * m;

        // w = C * e  (exploiting block-diagonal structure of C)
        const float q0 = fmaf(C00, e0, fmaf(C01, e1, C02 * e2));
        const float q1 = fmaf(C01, e0, fmaf(C00, e1, C02 * e2));
        const float q2 = fmaf(C20, e0 + e1, C22 * e2);
        const float q3 = CSH * e3;
        const float q4 = CSH * e4;
        const float q5 = CSH * e5;

        // A tiles (16x4 f32, ISA layout: lanes 0-15 hold K0/K1, 16-31 hold K2/K3;
        // row M = lane % 16 -> lane 16+j carries point j's K2/K3 components).
        v2f A1, A2, B1, B2;
        A1.x = lower ? e0 : e2;   A1.y = lower ? e1 : e3;
        A2.x = lower ? e4 : 0.0f; A2.y = lower ? e5 : 0.0f;
        // B tiles (4x16 f32): B[k][n] = w_k of point n, N = lane % 16.
        B1.x = lower ? q0 : q2;   B1.y = lower ? q1 : q3;
        B2.x = lower ? q4 : 0.0f; B2.y = lower ? q5 : 0.0f;

        // Q = ET(16x6) x W^T(6x16); diag(Q)[nn] = e_n . w_n = q_n exactly.
        v8f acc = {};
        acc = __builtin_amdgcn_wmma_f32_16x16x4_f32(
                  false, A1, false, B1, (short)0, acc, false, false);
        acc = __builtin_amdgcn_wmma_f32_16x16x4_f32(
                  false, A2, false, B2, (short)0, acc, false, false);

        // Diagonal lives at (VGPR r, lane r) for point r and
        // (VGPR r, lane 24+r) for point 8+r.  Accumulate q^2.
        #pragma unroll
        for (int r = 0; r < 8; ++r) {
            const bool sel = (lane == r) || (lane == 24 + r);
            const float t  = sel ? acc[r] : 0.0f;
            sumsq = fmaf(t, t, sumsq);
        }
        cnt += lower ? m : 0.0f;   // each point counted once (lower half)
    }

    // Wave-level tree reduction across the 32 lanes.
    #pragma unroll
    for (int off = 16; off >= 1; off >>= 1) {
        sumsq += __shfl_xor(sumsq, off, 32);
        cnt   += __shfl_xor(cnt,   off, 32);
    }
    if (lane == 0) {
        atomicAdd(&ws[0], sumsq);
        atomicAdd(&ws[1], cnt);
    }
}

__global__ void biomech_finalize(const float* __restrict__ ws,
                                 float* __restrict__ out)
{
    if (threadIdx.x == 0 && blockIdx.x == 0) {
        out[0] = sqrtf(ws[0]) / ws[1];
    }
}

extern "C" void kernel_launch(void* const* d_in, const int* in_sizes, int n_in,
                              void* d_out, int out_size, void* d_ws, size_t ws_size,
                              hipStream_t stream) {
    const float* gu  = (const float*)d_in[0];
    const float* gv  = (const float*)d_in[1];
    const float* gw  = (const float*)d_in[2];
    const float* sdf = (const float*)d_in[3];
    const int n = in_sizes[3];

    float* ws  = (float*)d_ws;
    float* out = (float*)d_out;

    // Zero the two accumulators on-stream (graph-capture safe, replay safe).
    biomech_init_ws<<<1, 32, 0, stream>>>(ws);

    // ~8K waves, grid-stride over 16-point tiles; purely HBM-bound (~160 MB).
    const int threads = 256;
    const int blocks  = 1024;
    biomech_wmma_kernel<<<blocks, threads, 0, stream>>>(gu, gv, gw, sdf, ws, n);

    biomech_finalize<<<1, 32, 0, stream>>>(ws, out);
}